// Encoder_M_22952305230093
// MI455X (gfx1250) — compile-verified
//
#include <hip/hip_runtime.h>
#include <math.h>
#include <stdint.h>

typedef __attribute__((ext_vector_type(2))) float v2f;
typedef __attribute__((ext_vector_type(8))) float v8f;

#define B_TOT   32768
#define D0      120
#define H0      4
#define ROW     960     // 120*4*2
#define P1D     30
#define P1ROW   240     // 30*4*2
#define XMROW   80      // 10*4*2
#define NLAT    100
#define K2      11

// flat output offsets (elements), return order: z, aug_xm, aug_x_sd, p1idx, p2idx
#define OFF_Z      ((size_t)0)
#define OFF_AUGXM  ((size_t)B_TOT * NLAT)
#define OFF_AUGSD  (OFF_AUGXM + (size_t)B_TOT * ROW)
#define OFF_P1IDX  (OFF_AUGSD + (size_t)B_TOT)
#define OFF_P2IDX  (OFF_P1IDX + (size_t)B_TOT * P1ROW)

// ---------------------------------------------------------------------------
// Kernel 1: one wave32 per batch row.
//   x1 row  --async DMA-->  LDS  --async DMA-->  aug_xm   (no VGPR round trip)
//   conv1+relu+pool4(+idx) and conv2+relu+pool3(+idx) computed out of LDS.
// Async global<->LDS ops are tracked with ASYNCcnt (ISA 10.x / CDNA5 ISA).
// ---------------------------------------------------------------------------
__global__ __launch_bounds__(256) void conv_pool_kernel(
    const float* __restrict__ x1,
    const float* __restrict__ w1, const float* __restrict__ b1,
    const float* __restrict__ w2, const float* __restrict__ b2,
    float* __restrict__ aug_xm, int* __restrict__ p1idx,
    int* __restrict__ p2idx, float* __restrict__ xm_ws)
{
    __shared__ float s_in[8][ROW];
    __shared__ float s_p1[8][P1ROW];
    const int lane  = threadIdx.x & 31;
    const int wv    = threadIdx.x >> 5;
    const int batch = blockIdx.x * 8 + wv;

    const float* xin  = x1     + (size_t)batch * ROW;
    float*       xout = aug_xm + (size_t)batch * ROW;

    // ---- async DMA: global row -> LDS (7 x b128 + 1 x b64 per lane) ----
    #pragma unroll
    for (int it = 0; it < 7; ++it) {
        const int fo = it * 128 + lane * 4;                 // float index
        unsigned lds = (unsigned)(uintptr_t)&s_in[wv][fo];  // LDS byte offset
        const float* g = xin + fo;
        asm volatile("global_load_async_to_lds_b128 %0, %1, off"
                     :: "v"(lds), "v"(g) : "memory");
    }
    {
        const int fo = 896 + lane * 2;
        unsigned lds = (unsigned)(uintptr_t)&s_in[wv][fo];
        const float* g = xin + fo;
        asm volatile("global_load_async_to_lds_b64 %0, %1, off"
                     :: "v"(lds), "v"(g) : "memory");
    }
    asm volatile("s_wait_asynccnt 0x0" ::: "memory");       // row resident in LDS

    // ---- echo aug_xm: async LDS -> global (copy bypasses VGPRs entirely) ----
    #pragma unroll
    for (int it = 0; it < 7; ++it) {
        const int fo = it * 128 + lane * 4;
        unsigned lds = (unsigned)(uintptr_t)&s_in[wv][fo];
        float* g = xout + fo;
        asm volatile("global_store_async_from_lds_b128 %0, %1, off"
                     :: "v"(g), "v"(lds) : "memory");
    }
    {
        const int fo = 896 + lane * 2;
        unsigned lds = (unsigned)(uintptr_t)&s_in[wv][fo];
        float* g = xout + fo;
        asm volatile("global_store_async_from_lds_b64 %0, %1, off"
                     :: "v"(g), "v"(lds) : "memory");
    }
    // stores drain on S_ENDPGM's implicit wait; LDS row is read-only below.

    float kw1[21], kw2[21];
    #pragma unroll
    for (int i = 0; i < 21; ++i) { kw1[i] = w1[i]; kw2[i] = w2[i]; }
    const float bb1 = b1[0], bb2 = b2[0];

    // ---- conv1 (7x3x1, pad 3,1,0) + relu + maxpool D:4 (first-max idx) ----
    for (int i = lane; i < P1ROW; i += 32) {
        const int w = i & 1, h = (i >> 1) & 3, dp = i >> 3;
        float best = -1e30f; int bo = 0;
        #pragma unroll
        for (int off = 0; off < 4; ++off) {
            const int d = dp * 4 + off;
            float acc = bb1;
            #pragma unroll
            for (int kd = 0; kd < 7; ++kd) {
                const int dd = d + kd - 3;
                if (dd < 0 || dd >= D0) continue;
                #pragma unroll
                for (int kh = 0; kh < 3; ++kh) {
                    const int hh = h + kh - 1;
                    if (hh < 0 || hh >= H0) continue;
                    acc += s_in[wv][(dd * 4 + hh) * 2 + w] * kw1[kd * 3 + kh];
                }
            }
            acc = fmaxf(acc, 0.0f);
            if (acc > best) { best = acc; bo = off; }
        }
        s_p1[wv][i] = best;
        p1idx[(size_t)batch * P1ROW + i] = ((dp * 4 + bo) * 4 + h) * 2 + w;
    }
    __syncthreads();

    // ---- conv2 (7x3x1 over D=30) + relu + maxpool D:3 (first-max idx) ----
    for (int i = lane; i < XMROW; i += 32) {
        const int w = i & 1, h = (i >> 1) & 3, dp = i >> 3;
        float best = -1e30f; int bo = 0;
        #pragma unroll
        for (int off = 0; off < 3; ++off) {
            const int d = dp * 3 + off;
            float acc = bb2;
            #pragma unroll
            for (int kd = 0; kd < 7; ++kd) {
                const int dd = d + kd - 3;
                if (dd < 0 || dd >= P1D) continue;
                #pragma unroll
                for (int kh = 0; kh < 3; ++kh) {
                    const int hh = h + kh - 1;
                    if (hh < 0 || hh >= H0) continue;
                    acc += s_p1[wv][(dd * 4 + hh) * 2 + w] * kw2[kd * 3 + kh];
                }
            }
            acc = fmaxf(acc, 0.0f);
            if (acc > best) { best = acc; bo = off; }
        }
        xm_ws[(size_t)batch * XMROW + i] = best;                 // xm [B,80]
        p2idx[(size_t)batch * XMROW + i] = ((dp * 3 + bo) * 4 + h) * 2 + w;
    }
}

// ---------------------------------------------------------------------------
// Kernel 2: head GEMMs on V_WMMA_F32_16X16X4_F32, one wave per 16 batch rows.
// All fragments fed from LDS: zero-padded weights staged once per block
// (kills the per-k-step exec-mask branches of round 1), A tiles staged with
// coalesced global reads. GEMM1 -> bias+ELU -> LDS transpose + sigmoid concat
// -> GEMM2 (K=12, 7 N-tiles) -> bias + 1/sqrt(1+eps) scale.
// ---------------------------------------------------------------------------
__global__ __launch_bounds__(256) void head_wmma_kernel(
    const float* __restrict__ xm_ws,
    const float* __restrict__ x2,
    const float* __restrict__ fcm1_w, const float* __restrict__ fcm1_b,
    const float* __restrict__ fc_w,   const float* __restrict__ fc_b,
    float* __restrict__ z, float* __restrict__ aug_sd)
{
    __shared__ float s_bw1[16][80];      // fcm1_w^T zero-padded: [n][k]
    __shared__ float s_bw2[112][12];     // fc_w zero-padded: [n][k]
    __shared__ float s_a1[8][16][80];    // per-wave A tile (xm rows)
    __shared__ float s_a2[8][16][12];    // concat [ELU(16x10) | sigmoid | 0]

    const int tid  = threadIdx.x;
    const int lane = tid & 31;
    const int wv   = tid >> 5;
    const int tile = blockIdx.x * 8 + wv;
    const int row0 = tile * 16;
    const int m    = lane & 15;
    const int half = lane >> 4;

    // block-wide staging of zero-padded B matrices
    for (int i = tid; i < 16 * 80; i += 256) {
        const int n = i / 80, k = i % 80;
        s_bw1[n][k] = (n < 10) ? fcm1_w[n * XMROW + k] : 0.0f;
    }
    for (int i = tid; i < 112 * 12; i += 256) {
        const int n = i / 12, k = i % 12;
        s_bw2[n][k] = (n < NLAT && k < K2) ? fc_w[n * K2 + k] : 0.0f;
    }
    // per-wave coalesced A-tile staging (16 rows x 80 = 1280 contiguous floats)
    {
        const float* asrc = xm_ws + (size_t)row0 * XMROW;
        float* adst = &s_a1[wv][0][0];
        for (int i = lane; i < 16 * XMROW; i += 32) adst[i] = asrc[i];
    }
    __syncthreads();

    // ---- GEMM1: [16x80] x [80x16]  (A lane<16 -> K{0,1}, lane>=16 -> K{2,3}) ----
    v8f c1 = {};
    #pragma unroll
    for (int kk = 0; kk < 20; ++kk) {
        const int k = kk * 4 + 2 * half;
        v2f a = *(const v2f*)&s_a1[wv][m][k];    // ds_load_b64, unconditional
        v2f b = *(const v2f*)&s_bw1[m][k];
        c1 = __builtin_amdgcn_wmma_f32_16x16x4_f32(
                 false, a, false, b, (short)0, c1, false, false);
    }

    // bias + ELU, scatter through LDS into A2 fragment layout (transpose)
    if (m < 10) {
        const float bias = fcm1_b[m];
        #pragma unroll
        for (int r = 0; r < 8; ++r) {
            float v = c1[r] + bias;
            v = v > 0.0f ? v : expm1f(v);
            a2_store: s_a2[wv][r + 8 * half][m] = v;   // row M = r+8*half, col m
        }
    }
    if (lane < 16) {
        const float xv = x2[row0 + lane];
        aug_sd[row0 + lane] = xv;                      // aug_x_sd = identity
        s_a2[wv][lane][10] = 1.0f / (1.0f + expf(-xv));
        s_a2[wv][lane][11] = 0.0f;                     // K pad to 12
    }
    __syncthreads();

    // ---- GEMM2: z = cat @ fc_w^T  (K=12, N tiled 7x16), scaled BN ----
    const float scale = rsqrtf(1.0f + 1e-5f);
    #pragma unroll
    for (int nt = 0; nt < 7; ++nt) {
        const int n = nt * 16 + m;
        v8f c2 = {};
        #pragma unroll
        for (int kk = 0; kk < 3; ++kk) {
            const int k = kk * 4 + 2 * half;
            v2f a = *(const v2f*)&s_a2[wv][m][k];
            v2f b = *(const v2f*)&s_bw2[n][k];
            c2 = __builtin_amdgcn_wmma_f32_16x16x4_f32(
                     false, a, false, b, (short)0, c2, false, false);
        }
        if (n < NLAT) {
            const float bias = fc_b[n];
            #pragma unroll
            for (int r = 0; r < 8; ++r) {
                z[(size_t)(row0 + r + 8 * half) * NLAT + n] = (c2[r] + bias) * scale;
            }
        }
    }
}

// ---------------------------------------------------------------------------
extern "C" void kernel_launch(void* const* d_in, const int* in_sizes, int n_in,
                              void* d_out, int out_size, void* d_ws, size_t ws_size,
                              hipStream_t stream)
{
    const float* x1     = (const float*)d_in[0];
    const float* x2     = (const float*)d_in[1];
    // d_in[2] = shifts, d_in[3] = nonzero_mask_xm: unused in eval forward
    const float* w1     = (const float*)d_in[4];
    const float* b1     = (const float*)d_in[5];
    const float* w2     = (const float*)d_in[6];
    const float* b2     = (const float*)d_in[7];
    const float* fcm1_w = (const float*)d_in[8];
    const float* fcm1_b = (const float*)d_in[9];
    const float* fc_w   = (const float*)d_in[10];
    const float* fc_b   = (const float*)d_in[11];

    float* out    = (float*)d_out;
    float* z      = out + OFF_Z;
    float* aug_xm = out + OFF_AUGXM;
    float* aug_sd = out + OFF_AUGSD;
    int*   p1     = (int*)(out + OFF_P1IDX);
    int*   p2     = (int*)(out + OFF_P2IDX);
    float* xm_ws  = (float*)d_ws;             // [B,80] staging, 10.5 MB

    conv_pool_kernel<<<B_TOT / 8, 256, 0, stream>>>(
        x1, w1, b1, w2, b2, aug_xm, p1, p2, xm_ws);

    head_wmma_kernel<<<(B_TOT / 16) / 8, 256, 0, stream>>>(
        xm_ws, x2, fcm1_w, fcm1_b, fc_w, fc_b, z, aug_sd);
}